// Net_11716670783492
// MI455X (gfx1250) — compile-verified
//
#include <hip/hip_runtime.h>
#include <math.h>

typedef __attribute__((ext_vector_type(2))) float v2f;
typedef __attribute__((ext_vector_type(8))) float v8f;

#define N_OSC 50
#define N_PER 50
#define STEPS 100
#define DTC   0.01f
#define LIMC  1.0471975511965976f   /* pi/3 */
#define GHALF 4.905f                /* m*g*l/2 */

// ---- LDS layout (floats) ----
#define OFF_ENC   0        /* 5000: enc[o][n][d]            */
#define OFF_BIAS  5000     /* 2500: osc_bias[o][n]          */
#define OFF_DEC   7500     /* 5000: dec[o][d][n]            */
#define OFF_FC4   12500    /* 2500: fc4_w                   */
#define OFF_A     15000    /* 2500: activations a[o][n]     */
#define OFF_H     17500    /* 128 : h                       */
#define OFF_H2    17628    /* 128 : h2                      */
#define OFF_S     17756    /* 100 : oscillator states       */
#define OFF_WSUM  17856    /* 8   : per-wave torque partial */
#define OFF_MISC  17864    /* 8   : [0]=direct [1]=fc4_b    */
#define LDS_FLOATS 17872

__global__ void cpg_limb_kernel(const float* __restrict__ x,
                                const float* __restrict__ fc1_w, const float* __restrict__ fc1_b,
                                const float* __restrict__ fc2_w, const float* __restrict__ fc2_b,
                                const float* __restrict__ fc3_w, const float* __restrict__ fc3_b,
                                const float* __restrict__ fcd_w, const float* __restrict__ fcd_b,
                                const float* __restrict__ fc4_w, const float* __restrict__ fc4_b,
                                const float* __restrict__ enc,   const float* __restrict__ osc_bias,
                                const float* __restrict__ dec,
                                float* __restrict__ out)
{
    extern __shared__ float lds[];
    const int tid  = threadIdx.x;          // 256 threads = 8 wave32
    const int wid  = tid >> 5;
    const int lane = tid & 31;

    // ---------------- stage weights into LDS (one-time) ----------------
    for (int i = tid; i < N_OSC * N_PER * 2; i += 256) lds[OFF_ENC + i] = enc[i];
    for (int i = tid; i < N_OSC * N_PER;     i += 256) lds[OFF_BIAS + i] = osc_bias[i];
    for (int i = tid; i < N_OSC * 2 * N_PER; i += 256) lds[OFF_DEC + i] = dec[i];
    for (int i = tid; i < N_OSC * N_PER;     i += 256) lds[OFF_FC4 + i] = fc4_w[i];
    if (tid == 0) lds[OFF_MISC + 1] = fc4_b[0];

    const float x0 = x[0], x1 = x[1];

    // fc1: h = relu(W1 @ x + b1)   (128x2 matvec, trivial)
    if (tid < 128) {
        float v = fmaf(fc1_w[tid * 2 + 0], x0, fmaf(fc1_w[tid * 2 + 1], x1, fc1_b[tid]));
        lds[OFF_H + tid] = fmaxf(v, 0.0f);
    }
    __syncthreads();

    // direct = fcd_w @ h + fcd_b   (wave 0, shuffle reduction)
    if (wid == 0) {
        float p = 0.0f;
        #pragma unroll
        for (int j = 0; j < 4; ++j) {
            int k = lane + 32 * j;
            p = fmaf(fcd_w[k], lds[OFF_H + k], p);
        }
        for (int off = 16; off > 0; off >>= 1) p += __shfl_down(p, off);
        if (lane == 0) lds[OFF_MISC + 0] = p + fcd_b[0];
    }

    // ---------------- fc2 via V_WMMA_F32_16X16X4_F32 ----------------
    // Each wave computes one 16-row tile of h2 = relu(W2 @ h + b2).
    // B broadcasts h across all 16 columns -> every result column equals W2@h.
    {
        const int r     = 16 * wid + (lane & 15);      // output row this lane feeds
        const int khalf = (lane >> 4) * 2;             // lanes 16-31 carry K+2,K+3
        v8f c = {};
        #pragma unroll 4
        for (int kk = 0; kk < 32; ++kk) {
            const int kb = 4 * kk + khalf;
            v2f a; a.x = fc2_w[r * 128 + kb]; a.y = fc2_w[r * 128 + kb + 1];
            v2f b; b.x = lds[OFF_H + kb];     b.y = lds[OFF_H + kb + 1];
            c = __builtin_amdgcn_wmma_f32_16x16x4_f32(false, a, false, b,
                                                      (short)0, c, false, false);
        }
        // C/D layout: VGPR i -> M=i (lanes 0-15) or M=8+i (lanes 16-31)
        if (lane == 0) {
            #pragma unroll
            for (int i = 0; i < 8; ++i) {
                int row = 16 * wid + i;
                lds[OFF_H2 + row] = fmaxf(c[i] + fc2_b[row], 0.0f);
            }
        } else if (lane == 16) {
            #pragma unroll
            for (int i = 0; i < 8; ++i) {
                int row = 16 * wid + 8 + i;
                lds[OFF_H2 + row] = fmaxf(c[i] + fc2_b[row], 0.0f);
            }
        }
    }
    __syncthreads();

    // ---------------- fc3 via V_WMMA_F32_16X16X4_F32 (100 rows, 7 tiles) ----------------
    if (wid < 7) {
        const int r     = 16 * wid + (lane & 15);
        const int rc    = (r < 100) ? r : 99;          // clamp to keep loads in-bounds
        const int khalf = (lane >> 4) * 2;
        v8f c = {};
        #pragma unroll 4
        for (int kk = 0; kk < 32; ++kk) {
            const int kb = 4 * kk + khalf;
            v2f a; a.x = fc3_w[rc * 128 + kb]; a.y = fc3_w[rc * 128 + kb + 1];
            v2f b; b.x = lds[OFF_H2 + kb];     b.y = lds[OFF_H2 + kb + 1];
            c = __builtin_amdgcn_wmma_f32_16x16x4_f32(false, a, false, b,
                                                      (short)0, c, false, false);
        }
        if (lane == 0) {
            #pragma unroll
            for (int i = 0; i < 8; ++i) {
                int row = 16 * wid + i;
                if (row < 100) lds[OFF_S + row] = c[i] + fc3_b[row];
            }
        } else if (lane == 16) {
            #pragma unroll
            for (int i = 0; i < 8; ++i) {
                int row = 16 * wid + 8 + i;
                if (row < 100) lds[OFF_S + row] = c[i] + fc3_b[row];
            }
        }
    }
    __syncthreads();

    // ---------------- 100-step scan (latency-bound; 2 barriers/step) ----------------
    float th = x0, om = 0.0f;                      // limb state lives in thread 255
    const float direct = lds[OFF_MISC + 0];
    const float fc4b   = lds[OFF_MISC + 1];

    for (int step = 0; step < STEPS; ++step) {
        // Phase A: a[o,n] = relu(enc.s + bias); fused torque partial
        float tq = 0.0f;
        #pragma unroll
        for (int it = 0; it < 10; ++it) {          // 10*256 > 2500
            int idx = tid + 256 * it;
            if (idx < N_OSC * N_PER) {
                int o = idx / N_PER;
                float v = fmaf(lds[OFF_ENC + 2 * idx + 0], lds[OFF_S + 2 * o + 0],
                          fmaf(lds[OFF_ENC + 2 * idx + 1], lds[OFF_S + 2 * o + 1],
                               lds[OFF_BIAS + idx]));
                v = fmaxf(v, 0.0f);
                lds[OFF_A + idx] = v;
                tq = fmaf(lds[OFF_FC4 + idx], v, tq);
            }
        }
        for (int off = 16; off > 0; off >>= 1) tq += __shfl_down(tq, off);
        if (lane == 0) lds[OFF_WSUM + wid] = tq;
        __syncthreads();

        // Phase B: oscillator state update (threads 0..99), limb update (thread 255)
        if (tid < 100) {
            const int o = tid >> 1, d = tid & 1;
            const float* dl = &lds[OFF_DEC + (o * 2 + d) * N_PER];
            const float* al = &lds[OFF_A + o * N_PER];
            float acc = 0.0f;
            #pragma unroll 5
            for (int n = 0; n < N_PER; ++n) acc = fmaf(dl[n], al[n], acc);
            lds[OFF_S + tid] += DTC * acc;
        } else if (tid == 255) {
            float torque = direct + fc4b;
            #pragma unroll
            for (int w = 0; w < 8; ++w) torque += lds[OFF_WSUM + w];
            float th2 = th + DTC * om;
            float om2 = om + DTC * (torque - GHALF * __sinf(th));
            bool hit = (th2 > LIMC) || (th2 < -LIMC);
            th2 = fminf(fmaxf(th2, -LIMC), LIMC);
            om2 = hit ? 0.0f : om2;
            out[2 * step + 0]     = th2;
            out[2 * step + 1]     = om2;
            out[2 * STEPS + step] = torque;
            th = th2; om = om2;
        }
        __syncthreads();
    }
}

extern "C" void kernel_launch(void* const* d_in, const int* in_sizes, int n_in,
                              void* d_out, int out_size, void* d_ws, size_t ws_size,
                              hipStream_t stream) {
    (void)in_sizes; (void)n_in; (void)d_ws; (void)ws_size; (void)out_size;
    const float* x      = (const float*)d_in[0];
    const float* fc1_w  = (const float*)d_in[1];
    const float* fc1_b  = (const float*)d_in[2];
    const float* fc2_w  = (const float*)d_in[3];
    const float* fc2_b  = (const float*)d_in[4];
    const float* fc3_w  = (const float*)d_in[5];
    const float* fc3_b  = (const float*)d_in[6];
    const float* fcd_w  = (const float*)d_in[7];
    const float* fcd_b  = (const float*)d_in[8];
    const float* fc4_w  = (const float*)d_in[9];
    const float* fc4_b  = (const float*)d_in[10];
    const float* enc    = (const float*)d_in[11];
    const float* oscb   = (const float*)d_in[12];
    const float* dec    = (const float*)d_in[13];

    cpg_limb_kernel<<<1, 256, LDS_FLOATS * sizeof(float), stream>>>(
        x, fc1_w, fc1_b, fc2_w, fc2_b, fc3_w, fc3_b,
        fcd_w, fcd_b, fc4_w, fc4_b, enc, oscb, dec, (float*)d_out);
}